// PatternMixer_5059471475274
// MI455X (gfx1250) — compile-verified
//
#include <hip/hip_runtime.h>
#include <math.h>

// Problem constants (from the reference)
#define NUM_BASIC 2
#define NUM_MIXED 4
#define NUM_NODES 7
#define CLIP_LEN  8
#define GROUP     7
#define NN        57            // N = 1 + 7*8
#define TT        15            // T = 2*8 - 1
#define TM        (TT * NUM_NODES)  // 105
#define MM_R      64            // mm rows padded to 4 WMMA tiles of 16
#define MM_C      60            // mm cols padded to 15 K-chunks of 4

typedef __attribute__((ext_vector_type(2))) float v2f;
typedef __attribute__((ext_vector_type(8))) float v8f;

__global__ __launch_bounds__(128) void pattern_mixer_kernel(
    const float* __restrict__ mat,     // [2][7][7]
    const float* __restrict__ te,      // [4][2][15]
    const float* __restrict__ mw,      // [4][2]
    const float* __restrict__ mb,      // [4]
    const float* __restrict__ mparam,  // [4][57][57]
    float* __restrict__ out)           // [4][57][57]
{
  const int ch  = blockIdx.x;          // mixed channel
  const int tid = threadIdx.x;

  __shared__ float sS[NUM_BASIC * TT];       // sigmoid(te)*mixer_w, per (h,t)
  __shared__ float sMixed[NUM_NODES * TM];   // mixed[n][t*7+m]
  __shared__ float sMM[MM_R * MM_C];         // padded mm tile for this channel
  __shared__ float sD[MM_R];                 // row sums -> d

  // ---- Phase A: per-(h,t) scale factors ----
  if (tid < NUM_BASIC * TT) {
    int h = tid / TT, t = tid % TT;
    float x  = te[ch * NUM_BASIC * TT + h * TT + t];
    float sg = 1.0f / (1.0f + expf(-x));
    sS[h * TT + t] = sg * mw[ch * NUM_BASIC + h];
  }
  __syncthreads();

  // ---- Phase B: mixed[n][t*7+m] = relu(b + sum_h sig(te)*w*mat) ----
  const float bi = mb[ch];
  for (int idx = tid; idx < NUM_NODES * TM; idx += 128) {
    int n = idx / TM, tm = idx % TM;
    int t = tm / NUM_NODES, m = tm % NUM_NODES;
    float v = bi
            + sS[t]      * mat[n * NUM_NODES + m]
            + sS[TT + t] * mat[NUM_NODES * NUM_NODES + n * NUM_NODES + m];
    sMixed[idx] = fmaxf(v, 0.0f);
  }
  __syncthreads();

  // ---- Phase C: build padded mm = param + clip slices, then mask ----
  for (int idx = tid; idx < MM_R * MM_C; idx += 128) {
    int r = idx / MM_C, c = idx % MM_C;
    float v = 0.0f;
    if (r < NN && c < NN) {
      v = mparam[ch * NN * NN + r * NN + c];
      if (r >= 1 && c >= 1) {
        int clip = (r - 1) / GROUP;
        int n    = (r - 1) % GROUP;
        int tm   = GROUP * (CLIP_LEN - 1 - clip) + (c - 1);
        v += sMixed[n * TM + tm];
      }
      if (r >= GROUP && (r % GROUP) == 0) {   // rows 7,14,...,56
        int  blk = r / GROUP - 1;
        bool own = (c >= 1 + GROUP * blk) && (c <= GROUP * (blk + 1));
        if (c >= 1 && (c % GROUP) != 0 && !own) v = 0.0f;
      }
    }
    sMM[idx] = v;
  }
  __syncthreads();

  // ---- Phase D: row sums via V_WMMA_F32_16X16X4_F32, B = ones ----
  // Wave w owns the 16-row tile [16w, 16w+16). A layout (fp32 16x4):
  // lanes 0-15 hold K={k0,k0+1}, lanes 16-31 hold K={k0+2,k0+3} for M=lane%16.
  {
    const int lane    = tid & 31;
    const int rowBase = (tid >> 5) * 16;
    const int row     = rowBase + (lane & 15);
    const int koff    = (lane >> 4) << 1;
    v8f acc = {0.f, 0.f, 0.f, 0.f, 0.f, 0.f, 0.f, 0.f};
    v2f bones; bones.x = 1.0f; bones.y = 1.0f;
    for (int k0 = 0; k0 < MM_C; k0 += 4) {
      v2f a;
      a.x = sMM[row * MM_C + k0 + koff];
      a.y = sMM[row * MM_C + k0 + koff + 1];
      acc = __builtin_amdgcn_wmma_f32_16x16x4_f32(
          /*neg_a=*/false, a, /*neg_b=*/false, bones,
          /*c_mod=*/(short)0, acc, /*reuse_a=*/false, /*reuse_b=*/false);
    }
    // C/D layout: VGPR v, lanes 0-15 -> M=v; lanes 16-31 -> M=8+v. Every N
    // column holds the same row sum; lanes with N==0 publish them.
    if ((lane & 15) == 0) {
      int mbase = rowBase + ((lane >> 4) << 3);
      #pragma unroll
      for (int v = 0; v < 8; ++v) sD[mbase + v] = acc[v];
    }
  }
  __syncthreads();

  // ---- d = max(rowsum, 1)^(-1/2) ----
  if (tid < MM_R) {
    float s = fmaxf(sD[tid], 1.0f);
    sD[tid] = 1.0f / sqrtf(s);
  }
  __syncthreads();

  // ---- Phase E: out = d[r] * mm * d[c] ----
  for (int idx = tid; idx < NN * NN; idx += 128) {
    int r = idx / NN, c = idx % NN;
    out[ch * NN * NN + idx] = sD[r] * sMM[r * MM_C + c] * sD[c];
  }
}

extern "C" void kernel_launch(void* const* d_in, const int* in_sizes, int n_in,
                              void* d_out, int out_size, void* d_ws, size_t ws_size,
                              hipStream_t stream) {
  const float* mat = (const float*)d_in[0];  // (2,7,7)
  const float* te  = (const float*)d_in[1];  // (4,2,15)
  const float* mw  = (const float*)d_in[2];  // (4,2)
  const float* mb  = (const float*)d_in[3];  // (4,)
  const float* mp  = (const float*)d_in[4];  // (4,57,57)
  float* out = (float*)d_out;                // (4,57,57)

  pattern_mixer_kernel<<<NUM_MIXED, 128, 0, stream>>>(mat, te, mw, mb, mp, out);
}